// WignerRotationZ_7232724927047
// MI455X (gfx1250) — compile-verified
//
#include <hip/hip_runtime.h>
#include <stdint.h>

typedef float v4f __attribute__((ext_vector_type(4)));

#define BLOCK 576          // 18 wave32s = 16 quads x 36 units
#define TILE_ELEMS 4096    // 64 rows x 64 channels (16 KB)

// Per-unit packed info: (|m| << 16) | (rowNeg << 8) | rowPos
// row(l,m) = l*l + l + m ; partner(l,m) = l*l + l - m
// Ordering: all 8 m==0 units first -> waves 0..3 take the identity path,
// waves 4..17 take the rotation path; zero intra-wave divergence.
#define U(rp, rn, mm) (((mm) << 16) | ((rn) << 8) | (rp))
__device__ __constant__ unsigned UNIT_INFO[36] = {
    // m == 0 rows (identity): l = 0..7, row = l*l + l
    U(0, 0, 0),  U(2, 2, 0),  U(6, 6, 0),  U(12, 12, 0),
    U(20, 20, 0), U(30, 30, 0), U(42, 42, 0), U(56, 56, 0),
    // l=1
    U(3, 1, 1),
    // l=2
    U(7, 5, 1),  U(8, 4, 2),
    // l=3
    U(13, 11, 1), U(14, 10, 2), U(15, 9, 3),
    // l=4
    U(21, 19, 1), U(22, 18, 2), U(23, 17, 3), U(24, 16, 4),
    // l=5
    U(31, 29, 1), U(32, 28, 2), U(33, 27, 3), U(34, 26, 4), U(35, 25, 5),
    // l=6
    U(43, 41, 1), U(44, 40, 2), U(45, 39, 3), U(46, 38, 4), U(47, 37, 5), U(48, 36, 6),
    // l=7
    U(57, 55, 1), U(58, 54, 2), U(59, 53, 3), U(60, 52, 4), U(61, 51, 5), U(62, 50, 6), U(63, 49, 7)
};

// CDNA5 async global->LDS copy, 16 bytes per active lane.
// NT hint: input is streamed exactly once -> don't pollute WGP$/L2.
__device__ __forceinline__ void async_copy_b128(uint64_t gaddr, uint32_t lds_addr) {
    asm volatile("global_load_async_to_lds_b128 %0, %1, off th:TH_LOAD_NT scope:SCOPE_SE"
                 :
                 : "v"(lds_addr), "v"(gaddr)
                 : "memory");
}

__device__ __forceinline__ void wait_async0() {
#if __has_builtin(__builtin_amdgcn_s_wait_asynccnt)
    __builtin_amdgcn_s_wait_asynccnt(0);
#else
    asm volatile("s_wait_asynccnt 0" ::: "memory");
#endif
}

__global__ __launch_bounds__(BLOCK)
void WignerRotationZ_kernel(const float* __restrict__ x,
                            const float* __restrict__ angle,
                            float* __restrict__ out, int N) {
    __shared__ float tile[TILE_ELEMS];

    const int n = blockIdx.x;
    if (n >= N) return;  // uniform across block

    const size_t slab = (size_t)n * TILE_ELEMS;
    const uint64_t gsrc = (uint64_t)(uintptr_t)(x + slab);
    // Low 32 bits of the flat address of a __shared__ object are the LDS byte offset.
    const uint32_t lds_base = (uint32_t)(uintptr_t)(&tile[0]);

    // Stage the whole 16 KB slab: 1024 b128 async transfers, fully coalesced,
    // straight-line (576 + 448 predicated) -- no loop bookkeeping.
    const unsigned tid = threadIdx.x;
    async_copy_b128(gsrc + (uint64_t)tid * 16u, lds_base + tid * 16u);
    if (tid < (TILE_ELEMS / 4 - BLOCK)) {  // tid < 448: whole-wave predicate
        const unsigned idx2 = tid + BLOCK;
        async_copy_b128(gsrc + (uint64_t)idx2 * 16u, lds_base + idx2 * 16u);
    }
    wait_async0();
    __syncthreads();

    const float phi = angle[n];

    const unsigned quad = tid & 15u;   // which float4 of the 64-channel row
    const unsigned unit = tid >> 4;    // 0..35

    const unsigned info = UNIT_INFO[unit];
    const unsigned rowP = info & 0xffu;
    const unsigned rowN = (info >> 8) & 0xffu;
    const unsigned m    = (info >> 16) & 0xffu;

    float* __restrict__ dst = out + slab;
    const unsigned cofs = quad * 4u;

    const v4f xP = *reinterpret_cast<const v4f*>(&tile[rowP * 64u + cofs]);

    if (m == 0u) {
        // identity rows (waves 0..3, uniform)
        __builtin_nontemporal_store(xP, reinterpret_cast<v4f*>(dst + rowP * 64u + cofs));
    } else {
        const v4f xN = *reinterpret_cast<const v4f*>(&tile[rowN * 64u + cofs]);
        float s, c;
        __sincosf((float)m * phi, &s, &c);
        const v4f oP = c * xP - s * xN;  // row with m > 0
        const v4f oN = c * xN + s * xP;  // partner row with m < 0
        __builtin_nontemporal_store(oP, reinterpret_cast<v4f*>(dst + rowP * 64u + cofs));
        __builtin_nontemporal_store(oN, reinterpret_cast<v4f*>(dst + rowN * 64u + cofs));
    }
}

extern "C" void kernel_launch(void* const* d_in, const int* in_sizes, int n_in,
                              void* d_out, int out_size, void* d_ws, size_t ws_size,
                              hipStream_t stream) {
    (void)n_in; (void)d_ws; (void)ws_size; (void)out_size;
    const float* x   = (const float*)d_in[0];   // (N, 64, 64) float32
    const float* ang = (const float*)d_in[1];   // (N,) float32
    float* out       = (float*)d_out;           // (N, 64, 64) float32
    const int N = in_sizes[1];                  // angle element count == N

    dim3 grid((unsigned)N), block(BLOCK);
    WignerRotationZ_kernel<<<grid, block, 0, stream>>>(x, ang, out, N);
}